// PixelmIoU_58574763983207
// MI455X (gfx1250) — compile-verified
//
#include <hip/hip_runtime.h>

// Pixel mIoU for pred[16,21,512,512] fp32 logits vs gt[16,1,512,512] int32.
// Memory-bound streaming reduction: argmax over 21 channels per pixel, then
// 3x21-bin integer histograms, then a tiny per-class IoU mean.
//
// MI455X reasoning: 352 MB pred + 16 MB gt read exactly once -> ~16 us floor
// at 23.3 TB/s. No matmul structure (WMMA inapplicable). Strategy: b128
// non-temporal streaming loads (data > 192 MB L2, zero reuse), wave32-aware
// two-level integer histogram (ds_add_u32 -> global_atomic_add_u32,
// bit-deterministic), single-wave32 shuffle finish.

namespace {
constexpr int NCLS  = 21;
constexpr int PLANE = 512 * 512;                       // 2^18 pixels per channel plane
constexpr int BATCH = 16;
constexpr long long NPIX = (long long)BATCH * PLANE;   // 4,194,304 pixels
constexpr int TPB = 256;                               // 8 wave32s per block
}

typedef float v4f __attribute__((ext_vector_type(4)));
typedef int   v4i __attribute__((ext_vector_type(4)));

// ws layout (uint32): [0..20]=inter, [21..41]=|pred==c|, [42..62]=|gt==c|
__global__ __launch_bounds__(64) void miou_zero(unsigned int* __restrict__ ws) {
  if (threadIdx.x < 3 * NCLS) ws[threadIdx.x] = 0u;
}

__global__ __launch_bounds__(TPB) void miou_hist(const float* __restrict__ pred,
                                                 const int*  __restrict__ gt,
                                                 unsigned int* __restrict__ ws) {
  __shared__ unsigned int h[3 * NCLS];
  const int t = threadIdx.x;
  if (t < 3 * NCLS) h[t] = 0u;
  __syncthreads();

  // Each thread owns 4 consecutive pixels -> b128 per-lane loads.
  const long long g = (long long)blockIdx.x * TPB + t;
  const long long P = g << 2;                 // first pixel of this group
  const int b = (int)(P >> 18);               // P / PLANE   (PLANE = 2^18)
  const int p = (int)(P & (PLANE - 1));       // P % PLANE   (groups never straddle planes)

  const float* base = pred + (long long)b * NCLS * PLANE + p;
  __builtin_prefetch(gt + P, 0, 0);           // gfx1250 global_prefetch_b8: warm gt line

  // Channel-0 seeds the running max; strict '>' keeps the FIRST maximal index
  // (matches jnp.argmax tie-breaking). NT hint: single-pass stream, bypass
  // cache retention (352 MB > 192 MB L2, zero reuse).
  v4f m = __builtin_nontemporal_load((const v4f*)base);
  int ax = 0, ay = 0, az = 0, aw = 0;
#pragma unroll
  for (int c = 1; c < NCLS; ++c) {
    const v4f v = __builtin_nontemporal_load((const v4f*)(base + (long long)c * PLANE));
    if (v.x > m.x) { m.x = v.x; ax = c; }
    if (v.y > m.y) { m.y = v.y; ay = c; }
    if (v.z > m.z) { m.z = v.z; az = c; }
    if (v.w > m.w) { m.w = v.w; aw = c; }
  }

  const v4i gv = __builtin_nontemporal_load((const v4i*)(gt + P));

  // Per-block histograms: no-return LDS atomics (ds_add_u32).
  atomicAdd(&h[NCLS + ax], 1u);
  atomicAdd(&h[NCLS + ay], 1u);
  atomicAdd(&h[NCLS + az], 1u);
  atomicAdd(&h[NCLS + aw], 1u);
  atomicAdd(&h[2 * NCLS + gv.x], 1u);
  atomicAdd(&h[2 * NCLS + gv.y], 1u);
  atomicAdd(&h[2 * NCLS + gv.z], 1u);
  atomicAdd(&h[2 * NCLS + gv.w], 1u);
  if (ax == gv.x) atomicAdd(&h[ax], 1u);
  if (ay == gv.y) atomicAdd(&h[ay], 1u);
  if (az == gv.z) atomicAdd(&h[az], 1u);
  if (aw == gv.w) atomicAdd(&h[aw], 1u);

  __syncthreads();
  // One flush per block per bin: global_atomic_add_u32 (integer => deterministic).
  if (t < 3 * NCLS) atomicAdd(&ws[t], h[t]);
}

__global__ __launch_bounds__(32) void miou_final(const unsigned int* __restrict__ ws,
                                                 float* __restrict__ out) {
  const int t = threadIdx.x;                  // exactly one wave32
  float iou = 0.0f;
  if (t < NCLS) {
    const float inter = (float)ws[t];
    const float cp    = (float)ws[NCLS + t];
    const float cg    = (float)ws[2 * NCLS + t];
    const float uni   = cp + cg - inter;      // union >= inter, so inter>0 => uni>0
    iou = (inter > 0.0f) ? (inter / uni) : 0.0f;
  }
#pragma unroll
  for (int off = 16; off > 0; off >>= 1)      // wave32 shuffle reduction
    iou += __shfl_down(iou, off, 32);
  if (t == 0) out[0] = iou * (1.0f / (float)NCLS);
}

extern "C" void kernel_launch(void* const* d_in, const int* in_sizes, int n_in,
                              void* d_out, int out_size, void* d_ws, size_t ws_size,
                              hipStream_t stream) {
  (void)in_sizes; (void)n_in; (void)out_size; (void)ws_size;
  const float*  pred = (const float*)d_in[0];
  const int*    gt   = (const int*)d_in[1];
  unsigned int* ws   = (unsigned int*)d_ws;
  float*        out  = (float*)d_out;

  miou_zero<<<1, 64, 0, stream>>>(ws);                      // re-init scratch every call
  const int groups = (int)(NPIX >> 2);                      // 1,048,576 pixel-quads
  miou_hist<<<groups / TPB, TPB, 0, stream>>>(pred, gt, ws);
  miou_final<<<1, 32, 0, stream>>>(ws, out);
}